// DendriteOutput_19602230739125
// MI455X (gfx1250) — compile-verified
//
#include <hip/hip_runtime.h>

// ---------------------------------------------------------------------------
// DendriteOutput: out[b,o] = sum_{d<32} x[b, o*32+d] * weight[o, o*32+d] + bias[o]
//   B=1024, OUT=2048, DPC=32, IN=65536. f32 in, f32 out.
//
// Memory-bound: x (256 MB, streamed once) dominates; ~11.3 us floor @ 23.3 TB/s.
// Formulated as a tiled GEMM with block-diagonal B so the f32 WMMA pipe
// (V_WMMA_F32_16X16X4_F32) performs the reduction at full f32 precision:
//   per 16b x 16o tile: D(16x16) = A(16x512) * B(512x16),
//   A = x tile (dense), B = block-diagonal weights (one nonzero column per
//   K=4 WMMA step). 128 chained WMMAs per tile; every D element is a result.
// ---------------------------------------------------------------------------

typedef __attribute__((ext_vector_type(2))) float v2f;
typedef __attribute__((ext_vector_type(8))) float v8f;

#define OUT_DIM 2048
#define DPC     32
#define BATCH   1024
#define IN_DIM  (OUT_DIM * DPC)          // 65536
#define OTILES  (OUT_DIM / 16)           // 128
#define BTILES  (BATCH / 16)             // 64
#define WAVES_PER_BLOCK 8

__global__ __launch_bounds__(256) void dendrite_wmma_kernel(
    const float* __restrict__ x,
    const float* __restrict__ weight,
    const float* __restrict__ bias,
    float* __restrict__ out)
{
    const int lane   = threadIdx.x & 31;   // wave32
    const int wave   = threadIdx.x >> 5;
    const int tileId = blockIdx.x * WAVES_PER_BLOCK + wave;
    // consecutive waves share an o-tile -> weight block reuse in L0/L2
    const int tile_b = (tileId & (BTILES - 1)) * 16;   // batch-tile origin
    const int tile_o = (tileId / BTILES) * 16;         // output-tile origin

    const int n = lane & 15;   // A: row M index; B/D: column N index
    const int h = lane >> 4;   // half-wave selector (K pairing per ISA layout)

    // ---- preload the 32 diagonal weights of output column (tile_o + n) ----
    // Lane (n,h) needs elements {4j+2h, 4j+2h+1 : j=0..7} of wdiag[o_n, 0..31].
    const int o_n = tile_o + n;
    const float* wrow = weight + (size_t)o_n * IN_DIM + (size_t)o_n * DPC;
    const v2f* wp2 = (const v2f*)wrow;
    v2f wreg[8];
#pragma unroll
    for (int j = 0; j < 8; ++j) wreg[j] = wp2[2 * j + h];

    // ---- per-lane x stream for the A fragments ----
    // ISA A(16x4 f32) layout: lane l holds A[l&15, k0+2h] , A[l&15, k0+2h+1]
    // => one aligned 8B load per WMMA step; 16B address stride per step.
    const v2f* xp2 =
        (const v2f*)(x + (size_t)(tile_b + n) * IN_DIM + (size_t)tile_o * DPC) + h;

    v8f acc = {};   // D/C accumulator, 16x16 f32

    // 512-deep reduction = 16 diagonal blocks (jb) x 8 K=4 steps (j).
    // B is nonzero only in column n == jb -> cndmask select, EXEC stays all-1s.
    for (int jb = 0; jb < 16; ++jb) {
#pragma unroll
        for (int j = 0; j < 8; ++j) {
            const int it = jb * 8 + j;
            v2f a = __builtin_nontemporal_load(&xp2[2 * it]);  // x: streamed, NT
            v2f b;
            b.x = (n == jb) ? wreg[j].x : 0.0f;
            b.y = (n == jb) ? wreg[j].y : 0.0f;
            // (neg_a, A, neg_b, B, c_mod, C, reuse_a, reuse_b)
            acc = __builtin_amdgcn_wmma_f32_16x16x4_f32(
                false, a, false, b, (short)0, acc, false, false);
        }
    }

    // ---- store D + bias ----
    // D layout: VGPR r, lanes 0-15 -> (M=r, N=lane); lanes 16-31 -> (M=r+8).
    const float bv = bias[o_n];
    float* orow = out + (size_t)(tile_b + h * 8) * OUT_DIM + o_n;
#pragma unroll
    for (int r = 0; r < 8; ++r) {
        orow[(size_t)r * OUT_DIM] = acc[r] + bv;
    }
}

extern "C" void kernel_launch(void* const* d_in, const int* in_sizes, int n_in,
                              void* d_out, int out_size, void* d_ws, size_t ws_size,
                              hipStream_t stream) {
    const float* x      = (const float*)d_in[0];
    const float* weight = (const float*)d_in[1];
    const float* bias   = (const float*)d_in[2];
    float* out          = (float*)d_out;

    const int tiles  = BTILES * OTILES;              // 8192 waves of work
    const int blocks = tiles / WAVES_PER_BLOCK;      // 1024 blocks x 256 thr
    dendrite_wmma_kernel<<<blocks, 256, 0, stream>>>(x, weight, bias, out);
}